// C3DLoss_57080115364097
// MI455X (gfx1250) — compile-verified
//
#include <hip/hip_runtime.h>
#include <hip/hip_bf16.h>
#include <math.h>

typedef float v2f __attribute__((ext_vector_type(2)));
typedef float v8f __attribute__((ext_vector_type(8)));

static constexpr int Bn = 4;
static constexpr int H  = 375;
static constexpr int W  = 1242;
static constexpr int HW = H * W;

// ---------------------------------------------------------------------------
// Kernel 1: zero the output (harness poisons d_out with 0xAA before timing).
// ---------------------------------------------------------------------------
__global__ void zero_out_kernel(float4* __restrict__ out4, float* __restrict__ out,
                                int n4, int n) {
    int i = blockIdx.x * blockDim.x + threadIdx.x;
    int stride = gridDim.x * blockDim.x;
    float4 z = make_float4(0.f, 0.f, 0.f, 0.f);
    for (int j = i; j < n4; j += stride) out4[j] = z;
    for (int j = n4 * 4 + i; j < n; j += stride) out[j] = 0.0f;
}

// ---------------------------------------------------------------------------
// Kernel 2: per-batch 6x4 transform matrices into d_ws.
//   rows 0..2 : T21[:3,:4]            (T21 = inv(Ts[tid]) @ Ts[b], rigid inverse)
//   rows 3..5 : K @ T21[:3,:4]        (u, v, w projection rows)
// Layout: ws[b*24 + r*4 + c], row-major.
// ---------------------------------------------------------------------------
__global__ void setup_mats_kernel(const float* __restrict__ Ts,
                                  const float* __restrict__ Kc,
                                  const int* __restrict__ seqn_p,
                                  float* __restrict__ M6) {
    int b = threadIdx.x;
    if (b >= Bn) return;
    int sn  = seqn_p[0];
    int tid = (b % sn == sn - 1) ? (b / sn) * sn : b + 1;
    const float* Ta = Ts + tid * 16;  // target frame (inverted)
    const float* Tb = Ts + b * 16;    // source frame

    float M[6][4];
    // T21 = [Ra^T | -Ra^T ta] @ [Rb | tb]  =>  R = Ra^T Rb, t = Ra^T (tb - ta)
    for (int r = 0; r < 3; ++r) {
        float a0 = Ta[0 * 4 + r], a1 = Ta[1 * 4 + r], a2 = Ta[2 * 4 + r]; // col r of Ra
        for (int c = 0; c < 3; ++c)
            M[r][c] = a0 * Tb[0 * 4 + c] + a1 * Tb[1 * 4 + c] + a2 * Tb[2 * 4 + c];
        M[r][3] = a0 * (Tb[3] - Ta[3]) + a1 * (Tb[7] - Ta[7]) + a2 * (Tb[11] - Ta[11]);
    }
    // projection rows: K @ M[0:3]
    for (int r = 0; r < 3; ++r) {
        const float* kr = Kc + b * 9 + r * 3;
        for (int c = 0; c < 4; ++c)
            M[3 + r][c] = kr[0] * M[0][c] + kr[1] * M[1][c] + kr[2] * M[2][c];
    }
    float* dst = M6 + b * 24;
    for (int r = 0; r < 6; ++r)
        for (int c = 0; c < 4; ++c)
            dst[r * 4 + c] = M[r][c];
}

// ---------------------------------------------------------------------------
// Kernel 3: WMMA reproject + scatter.
// One V_WMMA_F32_16X16X4_F32 per 16-pixel tile:
//   A (16x4): rows 0..5 = per-batch transform (loop invariant, 2 VGPRs/lane)
//     lane L: m = L&15; holds A[m][k0], A[m][k0+1] with k0 = (L<16) ? 0 : 2
//   B (4x16): col n = [x*d, y*d, d, 1] for pixel n
//     VGPR0: lanes 0-15 -> K=0 (x*d), lanes 16-31 -> K=2 (d)
//     VGPR1: lanes 0-15 -> K=1 (y*d), lanes 16-31 -> K=3 (1.0)
//   D (16x16): lane n (n<16) component r = row r applied to pixel n
//     => each of lanes 0..15 owns one pixel: tx,ty,tz,u,v,w in D[0..5]
// EXEC stays all-ones through the WMMA (selects are cndmask, uniform loop,
// clamped tail loads); predication only in the scatter phase.
// ---------------------------------------------------------------------------
__global__ void __launch_bounds__(256)
reproject_scatter_kernel(const float* __restrict__ depth,
                         const float* __restrict__ xy1,
                         const unsigned char* __restrict__ mask,
                         const int* __restrict__ seqn_p,
                         const float* __restrict__ M6,
                         float* __restrict__ out) {
    const int b    = blockIdx.y;
    const int lane = threadIdx.x & 31;
    const int wave = threadIdx.x >> 5;
    const int wavesPerBlock = blockDim.x >> 5;
    const int nTiles = (HW + 15) >> 4;

    const int sn   = seqn_p[0];
    const int tidb = (b % sn == sn - 1) ? (b / sn) * sn : b + 1;

    // Loop-invariant A fragment.
    const int m  = lane & 15;
    const int k0 = (lane < 16) ? 0 : 2;
    v2f a;
    a.x = (m < 6) ? M6[b * 24 + m * 4 + k0]     : 0.0f;
    a.y = (m < 6) ? M6[b * 24 + m * 4 + k0 + 1] : 0.0f;

    const float* dB = depth + (size_t)b * HW;
    const float* xB = xy1   + (size_t)b * 3 * HW;   // plane 0 = x, plane 1 = y
    const unsigned char* mB = mask + (size_t)b * HW;
    float* oB = out + (size_t)tidb * 3 * HW;

    const int tileStride = gridDim.x * wavesPerBlock;
    for (int tile = blockIdx.x * wavesPerBlock + wave; tile < nTiles; tile += tileStride) {
        const int p  = tile * 16 + m;
        const int pc = (p < HW) ? p : (HW - 1);      // clamp tail, no EXEC change

        const float d  = dB[pc];
        const float xv = xB[pc];
        const float yv = xB[HW + pc];

        v2f bm;
        bm.x = (lane < 16) ? xv * d : d;      // K=0 (x*d)  |  K=2 (z=d)
        bm.y = (lane < 16) ? yv * d : 1.0f;   // K=1 (y*d)  |  K=3 (1)

        v8f c = {0.f, 0.f, 0.f, 0.f, 0.f, 0.f, 0.f, 0.f};
        c = __builtin_amdgcn_wmma_f32_16x16x4_f32(
                /*neg_a=*/false, a, /*neg_b=*/false, bm,
                /*c_mod=*/(short)0, c, /*reuse_a=*/false, /*reuse_b=*/false);

        // Scatter phase (predicated; WMMA already retired its EXEC requirement).
        const bool active = (lane < 16) && (p < HW) && (mB[pc] != 0);
        const float w = c[5];                  // uvw[2]
        if (active && w > 0.0f) {
            const float inv = 1.0f / w;
            const float uf = rintf(c[3] * inv - 1.0f);   // round-half-even, like jnp.round
            const float vf = rintf(c[4] * inv - 1.0f);
            if (uf >= 0.0f && uf < (float)W && vf >= 0.0f && vf < (float)H) {
                const int idx = (int)vf * W + (int)uf;
                atomicAdd(&oB[idx],           c[0]);   // tx -> channel 0
                atomicAdd(&oB[HW + idx],      c[1]);   // ty -> channel 1
                atomicAdd(&oB[2 * HW + idx],  c[2]);   // tz -> channel 2
            }
        }
    }
}

// ---------------------------------------------------------------------------
// Launch
// ---------------------------------------------------------------------------
extern "C" void kernel_launch(void* const* d_in, const int* in_sizes, int n_in,
                              void* d_out, int out_size, void* d_ws, size_t ws_size,
                              hipStream_t stream) {
    const float*         depth = (const float*)d_in[0];
    const float*         xy1   = (const float*)d_in[1];
    const unsigned char* mask  = (const unsigned char*)d_in[2];
    const float*         Ts    = (const float*)d_in[3];
    const float*         Kc    = (const float*)d_in[4];
    const int*           seqn  = (const int*)d_in[5];
    float* out = (float*)d_out;
    float* ws  = (float*)d_ws;

    // 1) zero output
    const int n4 = out_size / 4;
    int zb = (n4 + 255) / 256;
    if (zb > 2048) zb = 2048;
    if (zb < 1) zb = 1;
    zero_out_kernel<<<zb, 256, 0, stream>>>((float4*)out, out, n4, out_size);

    // 2) per-batch transform matrices into workspace
    setup_mats_kernel<<<1, 32, 0, stream>>>(Ts, Kc, seqn, ws);

    // 3) WMMA reproject + scatter: grid-stride over 16-pixel tiles, 8 waves/block
    dim3 grid(512, Bn);
    reproject_scatter_kernel<<<grid, 256, 0, stream>>>(depth, xy1, mask, seqn, ws, out);
}